// Prompt_39599598469413
// MI455X (gfx1250) — compile-verified
//
#include <hip/hip_runtime.h>
#include <hip/hip_bf16.h>

typedef float v2f __attribute__((ext_vector_type(2)));
typedef float v4f __attribute__((ext_vector_type(4)));
typedef float v8f __attribute__((ext_vector_type(8)));

static constexpr int kB   = 512;           // batch
static constexpr int kD   = 1024;          // n_embd
static constexpr int kP   = 20;            // n_prompt (top-k)
static constexpr int kL   = 8;             // n_length
static constexpr int kNK  = 40;            // 2*n_prompt keys
static constexpr int kNKP = 48;            // padded to 3 WMMA N-tiles
static constexpr int kRow = kL * kD;       // 8192 floats per pool row

// ---------------------------------------------------------------------------
// Kernel 1: raw scores  S = Q · Kᵀ  via V_WMMA_F32_16X16X4_F32 (full f32).
// One wave (32 lanes) per 16 batch rows; 3 N-tiles cover the 40 keys.
// A layout (16x4 f32): lane m=l&15, VGPR0/1 = K = khalf, khalf+1 (khalf = 0|2).
// B layout (4x16):     lane n=l&15, VGPR0/1 = B[khalf..][n] = key[n][khalf..].
// C/D (16x16 f32):     VGPR v -> (M = v + (l<16?0:8), N = l&15).
// ---------------------------------------------------------------------------
__global__ __launch_bounds__(32) void att_wmma_kernel(
    const float* __restrict__ q, const float* __restrict__ kw,
    float* __restrict__ att /* [kB][kNKP] raw dot products */) {
  const int lane  = threadIdx.x;
  const int mbase = blockIdx.x << 4;
  const int m     = lane & 15;
  const int khalf = (lane >> 4) << 1;      // 0 or 2

  v8f c0 = {}, c1 = {}, c2 = {};
  const float* qrow  = q  + (size_t)(mbase + m) * kD;
  const float* krow0 = kw + (size_t)m * kD;
  const float* krow1 = kw + (size_t)(m + 16) * kD;
  const int   n2    = (m + 32 < kNK) ? (m + 32) : (kNK - 1);  // clamp, mask pad
  const float msk2  = (m + 32 < kNK) ? 1.0f : 0.0f;
  const float* krow2 = kw + (size_t)n2 * kD;

  for (int k0 = 0; k0 < kD; k0 += 4) {
    v2f a, b0, b1, b2;
    a.x  = qrow[k0 + khalf];          a.y  = qrow[k0 + khalf + 1];
    b0.x = krow0[k0 + khalf];         b0.y = krow0[k0 + khalf + 1];
    b1.x = krow1[k0 + khalf];         b1.y = krow1[k0 + khalf + 1];
    b2.x = krow2[k0 + khalf] * msk2;  b2.y = krow2[k0 + khalf + 1] * msk2;
    c0 = __builtin_amdgcn_wmma_f32_16x16x4_f32(false, a, false, b0, (short)0, c0, false, false);
    c1 = __builtin_amdgcn_wmma_f32_16x16x4_f32(false, a, false, b1, (short)0, c1, false, false);
    c2 = __builtin_amdgcn_wmma_f32_16x16x4_f32(false, a, false, b2, (short)0, c2, false, false);
  }

  const int rofs = (lane < 16) ? 0 : 8;
  #pragma unroll
  for (int v = 0; v < 8; ++v) {
    float* dst = att + (size_t)(mbase + v + rofs) * kNKP;
    dst[m]      = c0[v];
    dst[16 + m] = c1[v];
    dst[32 + m] = c2[v];
  }
}

// ---------------------------------------------------------------------------
// Kernel 2: squared L2 norms of query rows (blocks 0..511) and key rows
// (blocks 512..551). 256 threads reduce 1024 elements each.
// ---------------------------------------------------------------------------
__global__ __launch_bounds__(256) void norms_kernel(
    const float* __restrict__ q, const float* __restrict__ kw,
    float* __restrict__ normq, float* __restrict__ normk) {
  __shared__ float s[256];
  const int r = blockIdx.x;
  const float* row = (r < kB) ? (q + (size_t)r * kD) : (kw + (size_t)(r - kB) * kD);
  const int t = threadIdx.x;
  float acc = 0.0f;
  for (int i = t; i < kD; i += 256) { float v = row[i]; acc += v * v; }
  s[t] = acc;
  __syncthreads();
  for (int st = 128; st > 0; st >>= 1) {
    if (t < st) s[t] += s[t + st];
    __syncthreads();
  }
  if (t == 0) {
    if (r < kB) normq[r] = s[0];
    else        normk[r - kB] = s[0];
  }
}

// ---------------------------------------------------------------------------
// Kernel 3: per-row top-20 of 40 cosine scores (descending, lowest-index
// tie-break like jax.lax.top_k) + per-row loss partial sum(1 - cos).
// One thread per batch row; 40-element pool so selection-sort is trivial.
// ---------------------------------------------------------------------------
__global__ __launch_bounds__(256) void topk_loss_kernel(
    const float* __restrict__ att, const float* __restrict__ normq,
    const float* __restrict__ normk, int* __restrict__ idxout,
    float* __restrict__ rowloss) {
  const int b = blockIdx.x * blockDim.x + threadIdx.x;
  if (b >= kB) return;
  const float nq = normq[b];
  float cosv[kNK];
  #pragma unroll
  for (int n = 0; n < kNK; ++n) {
    float den = sqrtf(nq) * sqrtf(normk[n]);
    den = fmaxf(den, 1e-8f);
    cosv[n] = att[(size_t)b * kNKP + n] / den;
  }
  unsigned long long chosen = 0ull;
  float loss = 0.0f;
  for (int p = 0; p < kP; ++p) {
    float best = -__builtin_inff();
    int   bi   = 0;
    #pragma unroll
    for (int n = 0; n < kNK; ++n) {
      const bool free_slot = ((chosen >> n) & 1ull) == 0ull;
      if (free_slot && cosv[n] > best) { best = cosv[n]; bi = n; }
    }
    chosen |= (1ull << bi);
    idxout[(size_t)b * kP + p] = bi;
    loss += 1.0f - best;
  }
  rowloss[b] = loss;
}

// ---------------------------------------------------------------------------
// Kernel 4: deterministic reduction of 512 row losses -> scalar / (B*P).
// ---------------------------------------------------------------------------
__global__ __launch_bounds__(512) void reduce_loss_kernel(
    const float* __restrict__ rowloss, float* __restrict__ out_loss) {
  __shared__ float s[kB];
  const int t = threadIdx.x;
  s[t] = rowloss[t];
  __syncthreads();
  for (int st = kB / 2; st > 0; st >>= 1) {
    if (t < st) s[t] += s[t + st];
    __syncthreads();
  }
  if (t == 0) out_loss[0] = s[0] / (float)(kB * kP);
}

// ---------------------------------------------------------------------------
// Kernel 5: the bandwidth-dominant gather. One block per (b,p): copy one
// 32 KB pool row (8192 f32) with b128 loads (L2-hot: pool is only 1.3 MB)
// and non-temporal b128 stores (don't pollute L2 with the 335 MB stream).
// ---------------------------------------------------------------------------
__global__ __launch_bounds__(256) void gather_kernel(
    const float* __restrict__ pool, const int* __restrict__ idx,
    float* __restrict__ out) {
  const int bp  = blockIdx.x;                 // b * kP + p
  const int row = idx[bp];
  const v4f* __restrict__ src = (const v4f*)(pool + (size_t)row * kRow);
  v4f* __restrict__ dst       = (v4f*)(out + (size_t)bp * kRow);
  const int t = threadIdx.x;
  #pragma unroll
  for (int i = 0; i < kRow / 4 / 256; ++i) {   // 8 x b128 per thread
    v4f v = src[t + i * 256];
    __builtin_nontemporal_store(v, &dst[t + i * 256]);
  }
}

// ---------------------------------------------------------------------------
extern "C" void kernel_launch(void* const* d_in, const int* in_sizes, int n_in,
                              void* d_out, int out_size, void* d_ws, size_t ws_size,
                              hipStream_t stream) {
  (void)in_sizes; (void)n_in; (void)out_size; (void)ws_size;
  const float* q    = (const float*)d_in[0];   // [512,1024]
  const float* kw   = (const float*)d_in[1];   // [40,1024]
  const float* pool = (const float*)d_in[2];   // [40,8192]
  float* out = (float*)d_out;                  // emb flat + loss scalar

  // workspace layout (floats)
  float* ws      = (float*)d_ws;
  float* att     = ws;                          // kB*kNKP = 24576
  float* normq   = att + (size_t)kB * kNKP;     // 512
  float* normk   = normq + kB;                  // 48
  float* rowloss = normk + kNKP;                // 512
  int*   idx     = (int*)(rowloss + kB);        // kB*kP ints
  float* loss    = out + (size_t)kB * kP * kL * kD;  // last output element

  att_wmma_kernel  <<<kB / 16,        32, 0, stream>>>(q, kw, att);
  norms_kernel     <<<kB + kNK,      256, 0, stream>>>(q, kw, normq, normk);
  topk_loss_kernel <<<(kB + 255)/256, 256, 0, stream>>>(att, normq, normk, idx, rowloss);
  reduce_loss_kernel<<<1,             kB, 0, stream>>>(rowloss, loss);
  gather_kernel    <<<kB * kP,       256, 0, stream>>>(pool, idx, out);
}